// ModuleAMM_51625506898610
// MI455X (gfx1250) — compile-verified
//
#include <hip/hip_runtime.h>

#define N_    32
#define D_    576
#define NB_   1024
#define NPTS_ 16
#define SUBD_ 9
#define NSUB_ 64
#define M_    128

typedef __attribute__((ext_vector_type(2))) float v2f;
typedef __attribute__((ext_vector_type(4))) float v4f;
typedef __attribute__((ext_vector_type(8))) float v8f;

// Workspace layout (floats):
//   tab  : [NSUB][NPTS][M]   = 131072 f32 (512 KB), tab[(c*16+p)*128 + m]
//   ptsA : [NSUB][NPTS][12]  =  12288 f32 ( 48 KB), [2*pts(0..8), -||p||^2, 0, 0]

__global__ __launch_bounds__(256) void prep_tab(const float* __restrict__ pts,
                                                const float* __restrict__ w,
                                                float* __restrict__ tab) {
    int idx = blockIdx.x * 256 + threadIdx.x;   // 131072 total
    int m  = idx & (M_ - 1);
    int pc = idx >> 7;
    int p  = pc & (NPTS_ - 1);
    int c  = pc >> 4;
    const float* pr = pts + p * D_ + c * SUBD_;
    const float* wr = w   + m * D_ + c * SUBD_;
    float s = 0.f;
#pragma unroll
    for (int d = 0; d < SUBD_; ++d) s += pr[d] * wr[d];
    tab[(c * NPTS_ + p) * M_ + m] = s;
}

__global__ __launch_bounds__(256) void prep_ptsA(const float* __restrict__ pts,
                                                 float* __restrict__ ptsA) {
    int idx = blockIdx.x * 256 + threadIdx.x;   // 1024 total
    int p = idx & (NPTS_ - 1);
    int c = idx >> 4;
    const float* pr = pts + p * D_ + c * SUBD_;
    float* o = ptsA + (c * NPTS_ + p) * 12;
    float q = 0.f;
#pragma unroll
    for (int d = 0; d < SUBD_; ++d) {
        float v = pr[d];
        q += v * v;
        o[d] = 2.0f * v;
    }
    o[9]  = -q;    // folds -||p||^2 into the matmul via constant-1 row of B
    o[10] = 0.f;   // K=10,11 are hard zeros -> B operand there is don't-care
    o[11] = 0.f;
}

__device__ __forceinline__ float swap16_f(float v) {
    // ds_swizzle group-of-32: xor_mask=0x10, or=0, and=0x1f  (SWAPX16)
    return __builtin_bit_cast(float,
        __builtin_amdgcn_ds_swizzle(__builtin_bit_cast(int, v), 0x401F));
}
__device__ __forceinline__ int swap16_i(int v) {
    return __builtin_amdgcn_ds_swizzle(v, 0x401F);
}

__global__ __launch_bounds__(256) void amm_main(const float* __restrict__ x,
                                                const float* __restrict__ bias,
                                                const float* __restrict__ tab,
                                                const float* __restrict__ ptsA,
                                                float* __restrict__ out) {
    __shared__ float         ldsA[NSUB_ * NPTS_ * 12];   // 48 KB
    __shared__ unsigned char codes[NSUB_ * 128];         //  8 KB

    const int tid   = threadIdx.x;
    const int lane  = tid & 31;
    const int wv    = tid >> 5;                 // wave 0..7, owns b-subtile of 16
    const int n     = blockIdx.x >> 3;
    const int b0blk = (blockIdx.x & 7) * 128;

    // Stage ptsA once per workgroup (shared by all 8 waves, 64 reuses each).
    for (int i = tid; i < NSUB_ * NPTS_ * 12; i += 256) ldsA[i] = ptsA[i];
    __syncthreads();

    // ---------------- Phase 1: scores via WMMA f32 16x16x4, argmax -> codes
    const int col    = lane & 15;     // N column (b) / M row (p) within tile
    const int hi     = lane >> 4;     // lane-half selects K pair {0,1} vs {2,3}
    const int koff   = hi << 1;
    const int hibase = hi << 3;
    const int bloc   = wv * 16 + col;

    const float* xb = x + (size_t)n * D_ * NB_ + (b0blk + bloc);
    const float*   ar = ldsA + col * 12 + koff;   // += 192 per c
    const float*   xr = xb + (size_t)koff * NB_;  // += 9*NB per c
    const float*   x8 = xb + (size_t)8 * NB_;     // += 9*NB per c (in-bounds all lanes)
    unsigned char* cw = codes + bloc;             // += 128 per c

#pragma unroll 2
    for (int c = 0; c < NSUB_; ++c) {
        v2f a0 = {ar[0], ar[1]};
        v2f a1 = {ar[4], ar[5]};
        v2f a2 = {ar[8], ar[9]};                  // zeros for hi==1 (K=10,11)

        v2f b0 = {xr[0], xr[NB_]};
        v2f b1 = {xr[4 * NB_], xr[5 * NB_]};
        v2f b2 = {x8[0], 1.0f};                   // hi==1 half multiplies vs 0

        v8f acc = {};
        acc = __builtin_amdgcn_wmma_f32_16x16x4_f32(false, a0, false, b0, (short)0, acc, false, false);
        acc = __builtin_amdgcn_wmma_f32_16x16x4_f32(false, a1, false, b1, (short)0, acc, false, false);
        acc = __builtin_amdgcn_wmma_f32_16x16x4_f32(false, a2, false, b2, (short)0, acc, false, false);

        // Branchless argmax over the 8 p-values held by this lane-half.
        float best = acc[0];
        int   bp   = hibase;
#pragma unroll
        for (int r = 1; r < 8; ++r) {
            float v    = acc[r];
            bool  gt   = v > best;
            best       = gt ? v : best;
            bp         = gt ? hibase + r : bp;
        }
        // Branchless combine with partner half (lowest-p wins ties).
        float ob   = swap16_f(best);
        int   op   = swap16_i(bp);
        int   take = (int)(ob > best) | ((int)(ob == best) & (int)(op < bp));
        bp         = take ? op : bp;
        cw[0] = (unsigned char)bp;   // both halves hold identical winner

        ar += NPTS_ * 12;
        xr += SUBD_ * NB_;
        x8 += SUBD_ * NB_;
        cw += 128;
    }
    __syncthreads();

    // ---------------- Phase 2: out[n,:,b] = bias + sum_c tab[code,c,:]
    // Wave-uniform code per (b,c) => all 32 lanes read one contiguous 512B row.
    const v4f  bias4 = ((const v4f*)bias)[lane];
    const v4f* tab4  = (const v4f*)tab;

    for (int i = 0; i < 16; ++i) {
        const int bl = wv * 16 + i;
        const unsigned char* cr = codes + bl;
        v4f acc = bias4;
#pragma unroll 8
        for (int c = 0; c < NSUB_; ++c) {
            int code = cr[c * 128];
            acc += tab4[((c * NPTS_ + code) << 5) + lane];
        }
        float* op_ = out + ((size_t)n * M_ + lane * 4) * NB_ + (b0blk + bl);
        op_[0]       = acc.x;
        op_[NB_]     = acc.y;
        op_[2 * NB_] = acc.z;
        op_[3 * NB_] = acc.w;
    }
}

extern "C" void kernel_launch(void* const* d_in, const int* in_sizes, int n_in,
                              void* d_out, int out_size, void* d_ws, size_t ws_size,
                              hipStream_t stream) {
    const float* x      = (const float*)d_in[0];
    const float* pts    = (const float*)d_in[1];
    const float* weight = (const float*)d_in[2];
    const float* bias   = (const float*)d_in[3];
    // d_in[4] = temperature: only affects the soft path, which cancels in forward.
    float* out  = (float*)d_out;
    float* tab  = (float*)d_ws;                       // 512 KB
    float* ptsA = tab + NSUB_ * NPTS_ * M_;           //  48 KB

    prep_tab <<<(NSUB_ * NPTS_ * M_) / 256, 256, 0, stream>>>(pts, weight, tab);
    prep_ptsA<<<(NSUB_ * NPTS_) / 256,      256, 0, stream>>>(pts, ptsA);
    amm_main <<<N_ * (NB_ / 128),           256, 0, stream>>>(x, bias, tab, ptsA, out);
}